// GNNModel_2250562863480
// MI455X (gfx1250) — compile-verified
//
#include <hip/hip_runtime.h>

#define NNODES 100000
#define CH 128

typedef __attribute__((ext_vector_type(2))) float v2f;
typedef __attribute__((ext_vector_type(8))) float v8f;

// ---------------------------------------------------------------------------
// Degree count: one thread per edge, f32 atomic add (L2-resident, 400KB buf).
// ---------------------------------------------------------------------------
__global__ __launch_bounds__(256) void degree_kernel(const int* __restrict__ dst,
                                                     float* __restrict__ cnt, int E) {
    int e = blockIdx.x * blockDim.x + threadIdx.x;
    if (e < E) atomicAdd(&cnt[dst[e]], 1.0f);
}

// ---------------------------------------------------------------------------
// Scatter-add of feature rows: one wave per edge, each lane handles 4 channels
// via a coalesced float4 gather (full 512B row per wave) + 4 f32 atomics.
// Both source table and destination buffer fit in the 192MB L2.
// ---------------------------------------------------------------------------
__global__ __launch_bounds__(256) void scatter_kernel(const float* __restrict__ feat,
                                                      const int* __restrict__ src,
                                                      const int* __restrict__ dst,
                                                      float* __restrict__ agg, int E) {
    long long t = (long long)blockIdx.x * blockDim.x + threadIdx.x;
    int e = (int)(t >> 5);
    if (e >= E) return;
    int c4 = ((int)t & 31) * 4;          // lane's channel chunk
    int s = src[e];
    int d = dst[e];
    const float4 v = *(const float4*)(feat + (size_t)s * CH + c4);
    float* o = agg + (size_t)d * CH + c4;
    atomicAdd(o + 0, v.x);
    atomicAdd(o + 1, v.y);
    atomicAdd(o + 2, v.z);
    atomicAdd(o + 3, v.w);
}

// ---------------------------------------------------------------------------
// Fused SAGE linear: out = relu( (agg/max(cnt,1)) @ Wl^T + b + x @ Wr^T )
// One block = 16 output rows; 8 waves, each wave owns one 16x16 column tile.
// K-loop: 32 steps of V_WMMA_F32_16X16X4_F32 per weight matrix (64 WMMA/tile).
//
// Operand layouts (wave32, ISA 7.12.2):
//  A 16x4 f32:  lane L (L<16): {A[L][k],A[L][k+1]} ; L>=16: {A[L-16][k+2],A[L-16][k+3]}
//  B 4x16 f32:  lane L: {B[k+2*(L/16)][L%16], B[k+1+2*(L/16)][L%16]}  with B[k][j]=W[j][k]
//  => both are a float2 load at row (base + L%16), col (k + 2*(L/16)).
//  C/D 16x16:   VGPR r, lane L -> out[m0 + r + 8*(L/16)][j0 + L%16]
//
// `out` may alias `agg` (layer 2): all agg reads happen before the barrier,
// and a block only ever writes its own 16 rows.
// ---------------------------------------------------------------------------
__global__ __launch_bounds__(256) void sage_gemm_kernel(const float* __restrict__ agg,
                                                        const float* __restrict__ cnt,
                                                        const float* __restrict__ xin,
                                                        const float* __restrict__ Wl,
                                                        const float* __restrict__ Wr,
                                                        const float* __restrict__ bias,
                                                        float* __restrict__ out) {
    const int lane = threadIdx.x & 31;
    const int wv   = threadIdx.x >> 5;
    const int m0   = blockIdx.x * 16;      // 100000 / 16 = 6250 exactly
    const int j0   = wv * 16;
    const int lr   = lane & 15;
    const int hi   = lane >> 4;            // 0 or 1
    const int koff = hi * 2;

    const int rowA = m0 + lr;
    const float inv = 1.0f / fmaxf(cnt[rowA], 1.0f);   // fused mean
    const float* pAgg = agg + (size_t)rowA * CH + koff;
    const float* pX   = xin + (size_t)rowA * CH + koff;
    const float* pWl  = Wl  + (size_t)(j0 + lr) * CH + koff;
    const float* pWr  = Wr  + (size_t)(j0 + lr) * CH + koff;

    const float bv = bias[j0 + lr];        // fused bias (column j0+lr, all rows)
    v8f c;
#pragma unroll
    for (int r = 0; r < 8; ++r) c[r] = bv;

#pragma unroll
    for (int k = 0; k < CH; k += 4) {
        v2f aA = *(const v2f*)(pAgg + k);
        aA *= inv;
        v2f aX = *(const v2f*)(pX + k);
        v2f bL = *(const v2f*)(pWl + k);
        v2f bR = *(const v2f*)(pWr + k);
        c = __builtin_amdgcn_wmma_f32_16x16x4_f32(false, aA, false, bL,
                                                  (short)0, c, false, false);
        c = __builtin_amdgcn_wmma_f32_16x16x4_f32(false, aX, false, bR,
                                                  (short)0, c, false, false);
    }

    __syncthreads();   // allow out==agg aliasing: all reads done before writes

#pragma unroll
    for (int r = 0; r < 8; ++r) {
        const int row = m0 + r + (hi << 3);
        out[(size_t)row * CH + j0 + lr] = fmaxf(c[r], 0.0f);   // fused ReLU
    }
}

// ---------------------------------------------------------------------------
// Final 128 -> 2 projection: one wave per node, float4 per lane, shfl_xor
// tree reduction across the 32 lanes (wave32).
// ---------------------------------------------------------------------------
__global__ __launch_bounds__(256) void final_linear_kernel(const float* __restrict__ h,
                                                           const float* __restrict__ Wlin,
                                                           const float* __restrict__ blin,
                                                           float* __restrict__ out,
                                                           int n_nodes) {
    const int lane = threadIdx.x & 31;
    const int node = blockIdx.x * 8 + (threadIdx.x >> 5);
    if (node >= n_nodes) return;
    const float4 hv = *(const float4*)(h + (size_t)node * CH + lane * 4);
    const float4 w0 = *(const float4*)(Wlin + lane * 4);
    const float4 w1 = *(const float4*)(Wlin + CH + lane * 4);
    float d0 = hv.x * w0.x + hv.y * w0.y + hv.z * w0.z + hv.w * w0.w;
    float d1 = hv.x * w1.x + hv.y * w1.y + hv.z * w1.z + hv.w * w1.w;
#pragma unroll
    for (int off = 16; off > 0; off >>= 1) {
        d0 += __shfl_xor(d0, off, 32);
        d1 += __shfl_xor(d1, off, 32);
    }
    if (lane == 0) {
        out[(size_t)node * 2 + 0] = d0 + blin[0];
        out[(size_t)node * 2 + 1] = d1 + blin[1];
    }
}

// ---------------------------------------------------------------------------
// inputs: x, edge_index, W1_l, b1_l, W1_r, W2_l, b2_l, W2_r, W_lin, b_lin
// ws layout: cnt[N] | agg[N*128] | h1[N*128]   (~103 MB)
// ---------------------------------------------------------------------------
extern "C" void kernel_launch(void* const* d_in, const int* in_sizes, int n_in,
                              void* d_out, int out_size, void* d_ws, size_t ws_size,
                              hipStream_t stream) {
    const float* x    = (const float*)d_in[0];
    const int*   edge = (const int*)d_in[1];
    const int    E    = in_sizes[1] / 2;
    const int*   src  = edge;
    const int*   dst  = edge + E;
    const float* W1l  = (const float*)d_in[2];
    const float* b1l  = (const float*)d_in[3];
    const float* W1r  = (const float*)d_in[4];
    const float* W2l  = (const float*)d_in[5];
    const float* b2l  = (const float*)d_in[6];
    const float* W2r  = (const float*)d_in[7];
    const float* Wlin = (const float*)d_in[8];
    const float* blin = (const float*)d_in[9];
    float* out = (float*)d_out;

    const int    N  = NNODES;
    const size_t NF = (size_t)N * CH;
    float* cnt  = (float*)d_ws;
    float* agg  = cnt + N;        // N*4 = 400000 bytes, 16B-aligned
    float* h1   = agg + NF;

    const long long st = (long long)E * 32;
    const int scatter_blocks = (int)((st + 255) / 256);

    // --- degree (shared by both layers) ---
    hipMemsetAsync(cnt, 0, sizeof(float) * N, stream);
    degree_kernel<<<(E + 255) / 256, 256, 0, stream>>>(dst, cnt, E);

    // --- layer 1 ---
    hipMemsetAsync(agg, 0, sizeof(float) * NF, stream);
    scatter_kernel<<<scatter_blocks, 256, 0, stream>>>(x, src, dst, agg, E);
    sage_gemm_kernel<<<N / 16, 256, 0, stream>>>(agg, cnt, x, W1l, W1r, b1l, h1);

    // --- layer 2 (output aliases agg; safe per in-kernel barrier) ---
    hipMemsetAsync(agg, 0, sizeof(float) * NF, stream);
    scatter_kernel<<<scatter_blocks, 256, 0, stream>>>(h1, src, dst, agg, E);
    sage_gemm_kernel<<<N / 16, 256, 0, stream>>>(agg, cnt, h1, W2l, W2r, b2l, agg);

    // --- final projection 128 -> 2 ---
    final_linear_kernel<<<(N + 7) / 8, 256, 0, stream>>>(agg, Wlin, blin, out, N);
}